// PerAnnotatorModelForSequenceClassification_39556648796647
// MI455X (gfx1250) — compile-verified
//
#include <hip/hip_runtime.h>
#include <hip/hip_bf16.h>

// Problem constants (match reference)
#define AANN 256   // annotators
#define BB   64    // batch
#define DD   768   // hidden
#define LL   2     // labels
#define GG1  4
#define GG2  8

typedef __attribute__((ext_vector_type(2))) float v2f;
typedef __attribute__((ext_vector_type(8))) float v8f;

__device__ __forceinline__ v8f wmma_f32(v2f a, v2f b, v8f c) {
    // V_WMMA_F32_16X16X4_F32 : D(16x16 f32) = A(16x4) * B(4x16) + C
    return __builtin_amdgcn_wmma_f32_16x16x4_f32(
        /*neg_a=*/false, a, /*neg_b=*/false, b,
        /*c_mod=*/(short)0, c, /*reuse_a=*/false, /*reuse_b=*/false);
}

// One wave computes a 32(M) x 64(N) tile of C = A(64x768) @ B(768x768) + bias.
// Two A fragments (rows m0..m0+15 and m0+16..m0+31) share each B fragment,
// halving B-operand traffic vs a 16-row tile.
// Fragment layouts (ISA 7.12.2, 32-bit):
//   A 16x4 : lanes 0-15 hold M=lane, K={k,k+1}; lanes 16-31 hold K={k+2,k+3}.
//   B 4x16 : lanes 0-15 hold N=lane, K={k,k+1}; lanes 16-31 hold K={k+2,k+3}.
//   C/D    : VGPR j = row (j + 8*(lane>=16)), col = lane&15.
__device__ __forceinline__ void gemm_wave_32x64(
    const float* __restrict__ A,    // [64 x 768]
    const float* __restrict__ Bm,   // [768 x 768]
    const float* __restrict__ bias, // [768]
    float* __restrict__ C,          // [64 x 768]
    int m0, int n0)
{
    const int lane = threadIdx.x & 31;
    const int half = lane >> 4;     // which K-pair this lane supplies
    const int r    = lane & 15;     // M index (A) / N index (B, C)

    v8f acc[2][4];
    #pragma unroll
    for (int t = 0; t < 4; ++t) {
        const float bv = bias[n0 + t * 16 + r];
        #pragma unroll
        for (int j = 0; j < 8; ++j) { acc[0][t][j] = bv; acc[1][t][j] = bv; }
    }

    const float* arow0 = A + (size_t)(m0 + r) * DD;
    const float* arow1 = arow0 + (size_t)16 * DD;
    const float* bcol  = Bm + n0 + r;

    #pragma unroll 2
    for (int k = 0; k < DD; k += 4) {
        const int ka = k + 2 * half;
        v2f a0, a1;
        a0.x = arow0[ka];
        a0.y = arow0[ka + 1];
        a1.x = arow1[ka];
        a1.y = arow1[ka + 1];
        const float* bk = bcol + (size_t)ka * DD;
        #pragma unroll
        for (int t = 0; t < 4; ++t) {
            v2f b;
            b.x = bk[t * 16];
            b.y = bk[t * 16 + DD];
            acc[0][t] = wmma_f32(a0, b, acc[0][t]);
            acc[1][t] = wmma_f32(a1, b, acc[1][t]);
        }
    }

    float* crow = C + n0 + r;
    #pragma unroll
    for (int u = 0; u < 2; ++u) {
        #pragma unroll
        for (int t = 0; t < 4; ++t) {
            #pragma unroll
            for (int j = 0; j < 8; ++j) {
                const int row = m0 + u * 16 + j + 8 * half;
                crow[(size_t)row * DD + t * 16] = acc[u][t][j];
            }
        }
    }
}

// Stage 1: H1[g] = pooled @ W1[g] + b1[g], g in [0,4).
// 24 wave-tiles per GEMM (2 M-tiles of 32 x 12 N-tiles of 64) -> 3 blocks/GEMM.
__global__ __launch_bounds__(256) void stage1_kernel(
    const float* __restrict__ pooled, const float* __restrict__ W1,
    const float* __restrict__ b1, float* __restrict__ H1)
{
    const int g  = blockIdx.x / 3;
    const int wt = (blockIdx.x % 3) * 8 + (threadIdx.x >> 5); // 0..23
    const int m0 = (wt / 12) * 32;
    const int n0 = (wt % 12) * 64;
    gemm_wave_32x64(pooled,
                    W1 + (size_t)g * DD * DD,
                    b1 + (size_t)g * DD,
                    H1 + (size_t)g * BB * DD,
                    m0, n0);
}

// Stage 2: H2[p] = H1[p>>3] @ W2[p&7] + b2[p&7] for all 32 (g1,g2) pairs.
__global__ __launch_bounds__(256) void stage2_kernel(
    const float* __restrict__ H1, const float* __restrict__ W2,
    const float* __restrict__ b2, float* __restrict__ H2)
{
    const int p  = blockIdx.x / 3;
    const int g1 = p >> 3;
    const int g2 = p & 7;
    const int wt = (blockIdx.x % 3) * 8 + (threadIdx.x >> 5);
    const int m0 = (wt / 12) * 32;
    const int n0 = (wt % 12) * 64;
    gemm_wave_32x64(H1 + (size_t)g1 * BB * DD,
                    W2 + (size_t)g2 * DD * DD,
                    b2 + (size_t)g2 * DD,
                    H2 + (size_t)p * BB * DD,
                    m0, n0);
}

// Stage 3: logits[a] = H2[pair(a)] @ Wh[a] + bh[a]; output transposed to [B, A, L].
// One wave per (annotator, M-tile): N padded 2 -> 16; padded lanes carry zeros
// via in-bounds (r&1) addressing x 0 multiplier, so EXEC stays all-ones
// through every WMMA.
__global__ __launch_bounds__(256) void stage3_kernel(
    const float* __restrict__ H2, const float* __restrict__ Wh,
    const float* __restrict__ bh, const int* __restrict__ g1_idx,
    const int* __restrict__ g2_idx, float* __restrict__ out)
{
    const int wtile = blockIdx.x * 8 + (threadIdx.x >> 5); // 0..1023
    const int a  = wtile >> 2;                             // annotator
    const int m0 = (wtile & 3) * 16;                       // batch tile
    const int pair = g1_idx[a] * GG2 + g2_idx[a];

    const float* Am = H2 + (size_t)pair * BB * DD;
    const float* Bm = Wh + (size_t)a * DD * LL;

    const int lane = threadIdx.x & 31;
    const int half = lane >> 4;
    const int r    = lane & 15;
    const int rc   = r & 1;                    // always-in-bounds label index
    const float keep = (r < LL) ? 1.0f : 0.0f; // zero the padded columns

    v8f acc;
    const float bv = (r < LL) ? bh[a * LL + r] : 0.0f;
    #pragma unroll
    for (int j = 0; j < 8; ++j) acc[j] = bv;

    const float* arow = Am + (size_t)(m0 + r) * DD;

    #pragma unroll 4
    for (int k = 0; k < DD; k += 4) {
        const int ka = k + 2 * half;
        v2f a2;
        a2.x = arow[ka];
        a2.y = arow[ka + 1];
        v2f b;
        b.x = keep * Bm[ka * LL + rc];
        b.y = keep * Bm[(ka + 1) * LL + rc];
        acc = wmma_f32(a2, b, acc);
    }

    if (r < LL) {
        #pragma unroll
        for (int j = 0; j < 8; ++j) {
            const int row = m0 + j + 8 * half;                 // batch index
            out[(size_t)row * (AANN * LL) + a * LL + r] = acc[j];
        }
    }
}

extern "C" void kernel_launch(void* const* d_in, const int* in_sizes, int n_in,
                              void* d_out, int out_size, void* d_ws, size_t ws_size,
                              hipStream_t stream) {
    const float* pooled = (const float*)d_in[0];
    const float* W1     = (const float*)d_in[1];
    const float* b1     = (const float*)d_in[2];
    const float* W2     = (const float*)d_in[3];
    const float* b2     = (const float*)d_in[4];
    const float* Wh     = (const float*)d_in[5];
    const float* bh     = (const float*)d_in[6];
    const int*   g1     = (const int*)d_in[7];
    const int*   g2     = (const int*)d_in[8];
    float*       out    = (float*)d_out;

    // Workspace: H1 [4 x 64 x 768] then H2 [32 x 64 x 768]  (~7.1 MB total)
    float* H1 = (float*)d_ws;
    float* H2 = H1 + (size_t)GG1 * BB * DD;

    stage1_kernel<<<GG1 * 3, 256, 0, stream>>>(pooled, W1, b1, H1);
    stage2_kernel<<<GG1 * GG2 * 3, 256, 0, stream>>>(H1, W2, b2, H2);
    stage3_kernel<<<(AANN * 4) / 8, 256, 0, stream>>>(H2, Wh, bh, g1, g2, out);
}